// DeformDepthwiseSeparableConv_77472620085445
// MI455X (gfx1250) — compile-verified
//
#include <hip/hip_runtime.h>
#include <math.h>

typedef __attribute__((ext_vector_type(2))) float v2f;
typedef __attribute__((ext_vector_type(8))) float v8f;

#define BDIM 4
#define CDIM 64
#define OCDIM 128
#define HDIM 128
#define WDIM 128
#define HWDIM (HDIM*WDIM)      // 16384
#define NPIX (BDIM*HWDIM)      // 65536
#define KKPT 9

__device__ __forceinline__ int clampi(int v, int lo, int hi) {
    return v < lo ? lo : (v > hi ? hi : v);
}

// ---------- kernel 0: x NCHW -> NHWC (LDS-tiled transpose) ----------
__global__ void k_transpose(const float* __restrict__ x, float* __restrict__ xt) {
    __shared__ float tile[64][65];
    int b   = blockIdx.x >> 8;            // 256 hw-tiles of 64 per image
    int hw0 = (blockIdx.x & 255) * 64;
    int tid = threadIdx.x;
    #pragma unroll
    for (int i = 0; i < 16; ++i) {
        int lin = i * 256 + tid;
        int c = lin >> 6, j = lin & 63;   // consecutive tid -> consecutive j (coalesced read)
        tile[j][c] = x[((size_t)(b*CDIM + c)) * HWDIM + hw0 + j];
    }
    __syncthreads();
    #pragma unroll
    for (int i = 0; i < 16; ++i) {
        int lin = i * 256 + tid;
        int j = lin >> 6, c = lin & 63;   // consecutive tid -> consecutive c (coalesced write)
        xt[((size_t)(b*HWDIM + hw0 + j)) * CDIM + c] = tile[j][c];
    }
}

// ---------- kernel 1: pack conv weights -> Wre[kk][n(32)][c] ----------
__global__ void k_prepw(const float* __restrict__ ow, const float* __restrict__ mw,
                        float* __restrict__ wre) {
    int idx = blockIdx.x * blockDim.x + threadIdx.x;   // 9*32*64 = 18432
    if (idx >= 9*32*64) return;
    int c  = idx & 63;
    int n  = (idx >> 6) & 31;
    int kk = idx >> 11;
    float v = 0.f;
    if (n < 18)      v = ow[(n*CDIM + c)*KKPT + kk];
    else if (n < 27) v = mw[((n-18)*CDIM + c)*KKPT + kk];
    wre[idx] = v;
}

// ---------- kernel 2: offset/mask 3x3 conv as WMMA f32 GEMM ----------
// One wave: 16M x 32N tile (A fragment + border mask computed once, used by
// both n-accumulators). Fuses bias (offsets) and bias+sigmoid (mask) at store.
__global__ void k_convmask(const float* __restrict__ xt, const float* __restrict__ wre,
                           const float* __restrict__ ob, const float* __restrict__ mb,
                           float* __restrict__ comb) {
    int mt    = blockIdx.x * (blockDim.x >> 5) + (threadIdx.x >> 5); // 4096 m-tiles
    int lane  = threadIdx.x & 31;
    int mlane = lane & 15;
    int half  = lane >> 4;
    int m  = mt * 16 + mlane;       // tile of 16 pixels: same image, same row
    int b  = m >> 14;
    int hw = m & (HWDIM - 1);
    int y  = hw >> 7;
    int xx = hw & (WDIM - 1);
    int n0 = mlane;                 // D columns for this lane
    int n1 = 16 + mlane;

    v8f acc0 = {}, acc1 = {};
    #pragma unroll
    for (int kk = 0; kk < KKPT; ++kk) {
        int dy = kk / 3 - 1, dx = kk % 3 - 1;
        int yy = y + dy, xs = xx + dx;
        bool valid = (yy >= 0) && (yy < HDIM) && (xs >= 0) && (xs < WDIM);
        float vm = valid ? 1.f : 0.f;
        int yyc = clampi(yy, 0, HDIM-1), xsc = clampi(xs, 0, WDIM-1);
        const float* ap  = xt + ((size_t)((b*HDIM + yyc)*WDIM + xsc)) * CDIM + 2*half;
        const float* bp0 = wre + kk*2048 + n0*64 + 2*half;
        const float* bp1 = wre + kk*2048 + n1*64 + 2*half;
        #pragma unroll
        for (int c0 = 0; c0 < CDIM; c0 += 4) {
            v2f a = *(const v2f*)(ap + c0);
            a.x *= vm; a.y *= vm;
            v2f b0 = *(const v2f*)(bp0 + c0);
            v2f b1 = *(const v2f*)(bp1 + c0);
            acc0 = __builtin_amdgcn_wmma_f32_16x16x4_f32(
                false, a, false, b0, (short)0, acc0, false, false);
            acc1 = __builtin_amdgcn_wmma_f32_16x16x4_f32(
                false, a, false, b1, (short)0, acc1, false, false);
        }
    }

    // n0 in [0,16): always an offset channel. n1 in [16,32): offset/mask/pad.
    float bias0 = ob[n0];
    bool  isoff1  = (n1 < 18);
    bool  ismask1 = (n1 >= 18) && (n1 < 27);
    float bias1 = isoff1 ? ob[n1] : (ismask1 ? mb[n1-18] : 0.f);
    #pragma unroll
    for (int r = 0; r < 8; ++r) {
        int mr = mt*16 + r + 8*half;
        comb[(size_t)mr*32 + n0] = acc0[r] + bias0;
        float v1 = acc1[r] + bias1;
        if (ismask1) v1 = 1.f / (1.f + __expf(-v1));
        if (isoff1 || ismask1) comb[(size_t)mr*32 + n1] = v1;
    }
}

// ---------- kernel 3: deformable depthwise (bilinear, zero-pad) ----------
__device__ __forceinline__ float sample_nhwc(const float* __restrict__ xt,
                                             int b, int iy, int ix, int c) {
    if (iy < 0 || iy >= HDIM || ix < 0 || ix >= WDIM) return 0.f;
    return xt[((size_t)((b*HDIM + iy)*WDIM + ix)) * CDIM + c];
}

__global__ void k_deform(const float* __restrict__ xt, const float* __restrict__ comb,
                         const float* __restrict__ dw, float* __restrict__ dwout) {
    __shared__ int   s_iy[4][KKPT], s_ix[4][KKPT];
    __shared__ float s_w[4][KKPT][4];
    __shared__ float s_dw[CDIM*KKPT];          // 576 floats
    int tid = threadIdx.x;
    for (int i = tid; i < CDIM*KKPT; i += 256) s_dw[i] = dw[i];

    int p   = tid >> 6;               // pixel in block (0..3)
    int c   = tid & 63;               // channel
    int pix = blockIdx.x * 4 + p;
    int b   = pix >> 14;
    int hw  = pix & (HWDIM - 1);
    int y   = hw >> 7;
    int xx  = hw & (WDIM - 1);

    if (c < KKPT) {
        int kk = c;
        const float* cp = comb + (size_t)pix * 32;
        float dyo = cp[2*kk];
        float dxo = cp[2*kk + 1];
        float msk = cp[18 + kk];
        float py = (float)y  + (float)(kk/3 - 1) + dyo;
        float px = (float)xx + (float)(kk%3 - 1) + dxo;
        float y0f = floorf(py), x0f = floorf(px);
        float wy1 = py - y0f, wx1 = px - x0f;
        float wy0 = 1.f - wy1, wx0 = 1.f - wx1;
        s_iy[p][kk] = (int)y0f;
        s_ix[p][kk] = (int)x0f;
        s_w[p][kk][0] = msk * wy0 * wx0;
        s_w[p][kk][1] = msk * wy0 * wx1;
        s_w[p][kk][2] = msk * wy1 * wx0;
        s_w[p][kk][3] = msk * wy1 * wx1;
    }
    __syncthreads();

    float acc = 0.f;
    #pragma unroll
    for (int kk = 0; kk < KKPT; ++kk) {
        int iy0 = s_iy[p][kk], ix0 = s_ix[p][kk];
        float w00 = s_w[p][kk][0], w01 = s_w[p][kk][1];
        float w10 = s_w[p][kk][2], w11 = s_w[p][kk][3];
        float dwv = s_dw[c*KKPT + kk];
        float s00 = sample_nhwc(xt, b, iy0,   ix0,   c);
        float s01 = sample_nhwc(xt, b, iy0,   ix0+1, c);
        float s10 = sample_nhwc(xt, b, iy0+1, ix0,   c);
        float s11 = sample_nhwc(xt, b, iy0+1, ix0+1, c);
        acc += dwv * (w00*s00 + w01*s01 + w10*s10 + w11*s11);
    }
    dwout[(size_t)pix * CDIM + c] = acc;
}

// ---------- kernel 4: 1x1 conv as WMMA f32 GEMM, 32Mx32N tile/wave ----------
__global__ void k_pointwise(const float* __restrict__ a, const float* __restrict__ w,
                            const float* __restrict__ bias, float* __restrict__ out) {
    int wave  = blockIdx.x * (blockDim.x >> 5) + (threadIdx.x >> 5);
    int lane  = threadIdx.x & 31;
    int mt    = wave >> 2;           // 2048 m-tiles of 32
    int nt    = wave & 3;            // 4 n-tiles of 32
    int mlane = lane & 15;
    int half  = lane >> 4;
    int m0 = mt * 32 + mlane;
    int m1 = m0 + 16;
    int n0 = nt * 32 + mlane;
    int n1 = n0 + 16;

    const float* ap0 = a + (size_t)m0 * CDIM + 2*half;
    const float* ap1 = a + (size_t)m1 * CDIM + 2*half;
    const float* bp0 = w + (size_t)n0 * CDIM + 2*half;
    const float* bp1 = w + (size_t)n1 * CDIM + 2*half;

    v8f acc00 = {}, acc01 = {}, acc10 = {}, acc11 = {};
    #pragma unroll
    for (int k0 = 0; k0 < CDIM; k0 += 4) {
        v2f av0 = *(const v2f*)(ap0 + k0);
        v2f av1 = *(const v2f*)(ap1 + k0);
        v2f bv0 = *(const v2f*)(bp0 + k0);
        v2f bv1 = *(const v2f*)(bp1 + k0);
        acc00 = __builtin_amdgcn_wmma_f32_16x16x4_f32(
            false, av0, false, bv0, (short)0, acc00, false, false);
        acc01 = __builtin_amdgcn_wmma_f32_16x16x4_f32(
            false, av0, false, bv1, (short)0, acc01, false, false);
        acc10 = __builtin_amdgcn_wmma_f32_16x16x4_f32(
            false, av1, false, bv0, (short)0, acc10, false, false);
        acc11 = __builtin_amdgcn_wmma_f32_16x16x4_f32(
            false, av1, false, bv1, (short)0, acc11, false, false);
    }

    float bv0 = bias[n0], bv1 = bias[n1];
    #pragma unroll
    for (int r = 0; r < 8; ++r) {
        int mrA = mt*32 + r + 8*half;      // rows of acc00/acc01
        int mrB = mrA + 16;                // rows of acc10/acc11
        int bA = mrA >> 14, hwA = mrA & (HWDIM - 1);
        int bB = mrB >> 14, hwB = mrB & (HWDIM - 1);
        out[((size_t)(bA*OCDIM + n0)) * HWDIM + hwA] = acc00[r] + bv0;
        out[((size_t)(bA*OCDIM + n1)) * HWDIM + hwA] = acc01[r] + bv1;
        out[((size_t)(bB*OCDIM + n0)) * HWDIM + hwB] = acc10[r] + bv0;
        out[((size_t)(bB*OCDIM + n1)) * HWDIM + hwB] = acc11[r] + bv1;
    }
}

extern "C" void kernel_launch(void* const* d_in, const int* in_sizes, int n_in,
                              void* d_out, int out_size, void* d_ws, size_t ws_size,
                              hipStream_t stream) {
    const float* x  = (const float*)d_in[0];
    const float* ow = (const float*)d_in[1];
    const float* ob = (const float*)d_in[2];
    const float* mw = (const float*)d_in[3];
    const float* mb = (const float*)d_in[4];
    const float* dw = (const float*)d_in[5];
    const float* pw = (const float*)d_in[6];
    const float* pb = (const float*)d_in[7];
    float* out = (float*)d_out;

    char* ws = (char*)d_ws;
    size_t off = 0;
    float* xt    = (float*)(ws + off); off += (size_t)NPIX * CDIM * 4;  // 16 MB
    float* comb  = (float*)(ws + off); off += (size_t)NPIX * 32  * 4;   //  8 MB
    float* dwout = (float*)(ws + off); off += (size_t)NPIX * CDIM * 4;  // 16 MB
    float* wre   = (float*)(ws + off); off += 9 * 32 * 64 * 4;          // 72 KB

    // 0) x -> NHWC
    k_transpose<<<BDIM * (HWDIM/64), 256, 0, stream>>>(x, xt);
    // 1) pack 3x3 conv weights (offset+mask, N padded to 32)
    k_prepw<<<(9*32*64 + 255) / 256, 256, 0, stream>>>(ow, mw, wre);
    // 2) offset/mask conv via WMMA; 4096 waves (16Mx32N each), 4 waves/block
    k_convmask<<<1024, 128, 0, stream>>>(xt, wre, ob, mb, comb);
    // 3) deformable depthwise -> NHWC dwout
    k_deform<<<NPIX/4, 256, 0, stream>>>(xt, comb, dw, dwout);
    // 4) pointwise conv via WMMA; 8192 waves (32Mx32N each), 8 waves/block
    k_pointwise<<<1024, 256, 0, stream>>>(dwout, pw, pb, out);
}